// GraphEmbeddingNetwork_22368189677585
// MI455X (gfx1250) — compile-verified
//
#include <hip/hip_runtime.h>

typedef float v2f __attribute__((ext_vector_type(2)));
typedef float v8f __attribute__((ext_vector_type(8)));

#define NEG_INF (-3.402823e38f)

#if defined(__has_builtin)
#if __has_builtin(__builtin_amdgcn_global_load_async_to_lds_b32)
#define HAVE_ASYNC_LDS 1
#endif
#endif

__device__ __forceinline__ v8f wmma4(v2f a, v2f b, v8f c) {
  // D = A(16x4) * B(4x16) + C, f32 everywhere. 8-arg VOP3P form:
  // (neg_a, A, neg_b, B, c_mod, C, reuse_a, reuse_b)
  return __builtin_amdgcn_wmma_f32_16x16x4_f32(false, a, false, b, (short)0, c, false, false);
}
__device__ __forceinline__ float lrelu(float v) { return v >= 0.0f ? v : 0.2f * v; }

// ---------------- position embedding: x += disc^T pos_w + pos_b ----------------
__global__ void k_pos(const float* __restrict__ x, const float* __restrict__ disc,
                      const float* __restrict__ pw, const float* __restrict__ pb,
                      float* __restrict__ out) {
  int i = blockIdx.x * blockDim.x + threadIdx.x; // B*C*N = 16*128*1024
  if (i >= 16 * 128 * 1024) return;
  int n = i & 1023;
  int c = (i >> 10) & 127;
  int b = i >> 17;
  float acc = x[i] + pb[c];
#pragma unroll
  for (int j = 0; j < 6; ++j) acc += disc[(b * 6 + j) * 1024 + n] * pw[c * 6 + j];
  out[i] = acc;
}

// ---------------- per-point squared norms ----------------
__global__ void k_sq(const float* __restrict__ x, float* __restrict__ sq) {
  int i = blockIdx.x * blockDim.x + threadIdx.x; // B*N
  if (i >= 16 * 1024) return;
  int n = i & 1023, b = i >> 10;
  const float* xb = x + b * 128 * 1024 + n;
  float s = 0.0f;
  for (int c = 0; c < 128; ++c) { float v = xb[c * 1024]; s += v * v; }
  sq[i] = s;
}

// ---------------- fused KNN: WMMA X^T X stripe in LDS + iterative top-10 ----------------
__global__ __launch_bounds__(128) void k_knn(const float* __restrict__ x,
                                             const float* __restrict__ sq,
                                             int* __restrict__ idx) {
  __shared__ float dist[16 * 1024]; // 64 KB stripe: 16 rows x 1024 cols
  const int b = blockIdx.y;
  const int row0 = blockIdx.x * 16;
  const int lane = threadIdx.x & 31;
  const int wv = threadIdx.x >> 5; // 4 waves
  const int m16 = lane & 15;
  const int h = lane >> 4;
  const float* xb = x + b * 128 * 1024;
  const float* sqb = sq + b * 1024;

  // Preload A fragments (16 rows x 128 channels) -> 32 K-steps of 4
  v2f afrag[32];
#pragma unroll
  for (int s = 0; s < 32; ++s) {
    int c = 4 * s + 2 * h;
    afrag[s].x = xb[c * 1024 + row0 + m16];
    afrag[s].y = xb[(c + 1) * 1024 + row0 + m16];
  }
  float sqrow[8];
#pragma unroll
  for (int i = 0; i < 8; ++i) sqrow[i] = sqb[row0 + 8 * h + i];

  for (int t = wv; t < 64; t += 4) {
    int col0 = t * 16;
    v8f acc = {0.f, 0.f, 0.f, 0.f, 0.f, 0.f, 0.f, 0.f};
#pragma unroll
    for (int s = 0; s < 32; ++s) {
      int c = 4 * s + 2 * h;
      v2f bf;
      bf.x = xb[c * 1024 + col0 + m16];
      bf.y = xb[(c + 1) * 1024 + col0 + m16];
      acc = wmma4(afrag[s], bf, acc);
    }
    float sqc = sqb[col0 + m16];
#pragma unroll
    for (int i = 0; i < 8; ++i)
      dist[(8 * h + i) * 1024 + col0 + m16] = 2.0f * acc[i] - sqrow[i] - sqc;
  }
  __syncthreads();

  // top-10 per row; wave wv owns rows {wv, wv+4, wv+8, wv+12}
  for (int rr = wv; rr < 16; rr += 4) {
    float* drow = dist + rr * 1024;
    for (int t = 0; t < 10; ++t) {
      float best = NEG_INF;
      int bi = 1 << 30;
      for (int j = lane; j < 1024; j += 32) {
        float v = drow[j];
        if (v > best) { best = v; bi = j; }
      }
#pragma unroll
      for (int off = 16; off > 0; off >>= 1) {
        float ov = __shfl_xor(best, off, 32);
        int oi = __shfl_xor(bi, off, 32);
        if (ov > best || (ov == best && oi < bi)) { best = ov; bi = oi; }
      }
      if (lane == 0) {
        idx[(b * 1024 + row0 + rr) * 10 + t] = bi;
        drow[bi] = NEG_INF;
      }
      __syncthreads(); // uniform trip counts: safe; publishes the -inf poke
    }
  }
}

// ---------------- edge GEMM: y[(bnk), o] = f[(bnk), c] * w[o, c]; channel stats via atomics --
__global__ __launch_bounds__(256) void k_edge_gemm(
    const float* __restrict__ x, const int* __restrict__ idx,
    const float* __restrict__ w, const float* __restrict__ para, int pb,
    float* __restrict__ y, float* __restrict__ stats, int O) {
  __shared__ float fT[16 * 256]; // 16 rows x 256 edge-feature channels
  const int rb = blockIdx.x;     // 163840/16 = 10240 row blocks
  const int tid = threadIdx.x;
  const int lane = tid & 31, wv = tid >> 5; // 8 waves
  const float p0 = para[4 * pb], p1 = para[4 * pb + 1];

  { // gather the 16x256 feature tile: row r = (b,n,k); c<128 diff*p0, else center*p1
    const int r = tid & 15;
    const int c0 = (tid >> 4) * 16;
    const int gr = rb * 16 + r;
    const int b = gr / 10240; // N*K
    const int rem = gr - b * 10240;
    const int n = rem / 10;
    const int k = rem - n * 10;
    const int j = idx[(b * 1024 + n) * 10 + k];
    const float* xb = x + b * 128 * 1024;
#pragma unroll
    for (int cc = 0; cc < 16; ++cc) {
      int c = c0 + cc;
      float v;
      if (c < 128) v = (xb[c * 1024 + j] - xb[c * 1024 + n]) * p0;
      else         v = xb[(c - 128) * 1024 + n] * p1;
      fT[r * 256 + c] = v;
    }
  }
  __syncthreads();

  const int m16 = lane & 15, h = lane >> 4;
  const int ntiles = O >> 4;
  for (int t = wv; t < ntiles; t += 8) {
    const int o = t * 16 + m16;
    v8f acc = {0.f, 0.f, 0.f, 0.f, 0.f, 0.f, 0.f, 0.f};
#pragma unroll 8
    for (int s = 0; s < 64; ++s) {
      int c = 4 * s + 2 * h;
      v2f a = *(const v2f*)&fT[m16 * 256 + c];
      v2f bf = *(const v2f*)&w[o * 256 + c];
      acc = wmma4(a, bf, acc);
    }
#pragma unroll
    for (int i = 0; i < 8; ++i)
      y[(rb * 16 + 8 * h + i) * O + o] = acc[i];
    if (stats) {
      float s1 = 0.f, s2 = 0.f;
#pragma unroll
      for (int i = 0; i < 8; ++i) { s1 += acc[i]; s2 += acc[i] * acc[i]; }
      s1 += __shfl_xor(s1, 16, 32);
      s2 += __shfl_xor(s2, 16, 32);
      if (h == 0) { atomicAdd(&stats[o], s1); atomicAdd(&stats[O + o], s2); }
    }
  }
}

// ---------------- fold BN stats into (scale, shift) in place ----------------
__global__ void k_bnparams(float* __restrict__ stats, const float* __restrict__ g,
                           const float* __restrict__ bta, float invM, int O) {
  int o = blockIdx.x * blockDim.x + threadIdx.x;
  if (o >= O) return;
  float mean = stats[o] * invM;
  float var = stats[O + o] * invM - mean * mean;
  float sc = g[o] * rsqrtf(var + 1e-5f);
  stats[o] = sc;
  stats[O + o] = bta[o] - mean * sc;
}

// ---------------- edge finalize: lrelu(bn(y)) [+ res] -> max over k, transpose ----------------
__global__ void k_edge_fin(const float* __restrict__ y, const float* __restrict__ r,
                           const float* __restrict__ stats, float* __restrict__ out, int O) {
  int i = blockIdx.x * blockDim.x + threadIdx.x; // B*N*O, o fastest (coalesced reads)
  if (i >= 16 * 1024 * O) return;
  int o = i % O;
  int bn = i / O;
  float sc = stats[o], sh = stats[O + o];
  int base = bn * 10 * O + o;
  float m = NEG_INF;
#pragma unroll
  for (int k = 0; k < 10; ++k) {
    float v = lrelu(y[base + k * O] * sc + sh);
    if (r) v += r[base + k * O];
    m = fmaxf(m, v);
  }
  int b = bn >> 10, n = bn & 1023;
  out[(b * O + o) * 1024 + n] = m;
}

// ---------------- w5: y5[b,o,n] = concat(x1..x4)[b,c,n] * w5[o,c], 640->1024 ----------------
__device__ __forceinline__ const float* xc_src(const float* x1, const float* x2,
                                               const float* x3, const float* x4,
                                               int b, int c, int n) {
  if (c < 128) return &x1[(b * 128 + c) * 1024 + n];
  if (c < 256) return &x2[(b * 128 + (c - 128)) * 1024 + n];
  if (c < 384) return &x3[(b * 128 + (c - 256)) * 1024 + n];
  return &x4[(b * 256 + (c - 384)) * 1024 + n];
}

__global__ __launch_bounds__(256) void k_w5_gemm(
    const float* __restrict__ x1, const float* __restrict__ x2,
    const float* __restrict__ x3, const float* __restrict__ x4,
    const float* __restrict__ w5, float* __restrict__ y5, float* __restrict__ stats) {
  __shared__ float fT[16 * 640]; // 40 KB tile
  const int rb = blockIdx.x; // 1024 row blocks of 16 points
  const int b = rb >> 6;
  const int n0 = (rb & 63) << 4;
  const int tid = threadIdx.x;
  const int lane = tid & 31, wv = tid >> 5;
  { // stage the 16x640 activation tile through the gfx1250 async global->LDS path
    const int r = tid & 15;
#if HAVE_ASYNC_LDS
    typedef __attribute__((address_space(1))) int g_i32;
    typedef __attribute__((address_space(3))) int l_i32;
    for (int c = tid >> 4; c < 640; c += 16) {
      const float* gp = xc_src(x1, x2, x3, x4, b, c, n0 + r);
      float* lp = &fT[r * 640 + c];
      __builtin_amdgcn_global_load_async_to_lds_b32(
          (g_i32*)(uintptr_t)gp, (l_i32*)(uintptr_t)lp, 0, 0);
    }
    asm volatile("s_wait_asynccnt 0x0" ::: "memory");
#else
    for (int c = tid >> 4; c < 640; c += 16)
      fT[r * 640 + c] = *xc_src(x1, x2, x3, x4, b, c, n0 + r);
#endif
  }
  __syncthreads();
  const int m16 = lane & 15, h = lane >> 4;
  for (int t = wv; t < 64; t += 8) {
    const int o = t * 16 + m16;
    v8f acc = {0.f, 0.f, 0.f, 0.f, 0.f, 0.f, 0.f, 0.f};
#pragma unroll 8
    for (int s = 0; s < 160; ++s) {
      int c = 4 * s + 2 * h;
      v2f a = *(const v2f*)&fT[m16 * 640 + c];
      v2f bf = *(const v2f*)&w5[o * 640 + c];
      acc = wmma4(a, bf, acc);
    }
#pragma unroll
    for (int i = 0; i < 8; ++i)
      y5[(b * 1024 + o) * 1024 + n0 + 8 * h + i] = acc[i];
    float s1 = 0.f, s2 = 0.f;
#pragma unroll
    for (int i = 0; i < 8; ++i) { s1 += acc[i]; s2 += acc[i] * acc[i]; }
    s1 += __shfl_xor(s1, 16, 32);
    s2 += __shfl_xor(s2, 16, 32);
    if (h == 0) { atomicAdd(&stats[o], s1); atomicAdd(&stats[1024 + o], s2); }
  }
}

// ---------------- pool: per (b,o) wave computes max & mean over n of lrelu(bn(y5)) --------
__global__ __launch_bounds__(256) void k_pool(const float* __restrict__ y5,
                                              const float* __restrict__ stats,
                                              float* __restrict__ p) {
  int gw = blockIdx.x * 8 + (threadIdx.x >> 5); // 16384 waves
  int lane = threadIdx.x & 31;
  int b = gw >> 10, o = gw & 1023;
  float sc = stats[o], sh = stats[1024 + o];
  const float* row = y5 + (b * 1024 + o) * 1024;
  float mx = NEG_INF, sm = 0.f;
  for (int n = lane; n < 1024; n += 32) {
    float v = lrelu(row[n] * sc + sh);
    mx = fmaxf(mx, v);
    sm += v;
  }
#pragma unroll
  for (int off = 16; off > 0; off >>= 1) {
    mx = fmaxf(mx, __shfl_xor(mx, off, 32));
    sm += __shfl_xor(sm, off, 32);
  }
  if (lane == 0) {
    p[b * 2048 + o] = mx;
    p[b * 2048 + 1024 + o] = sm * (1.0f / 1024.0f);
  }
}

// ---------------- small FC: out[b,o] = in[b,:] . wt[o,:] ----------------
__global__ void k_lin(const float* __restrict__ in, const float* __restrict__ wt,
                      float* __restrict__ out, int Cin, int Cout) {
  int i = blockIdx.x * blockDim.x + threadIdx.x;
  if (i >= 16 * Cout) return;
  int o = i % Cout, b = i / Cout;
  const float* wr = wt + o * Cin;
  const float* ir = in + b * Cin;
  float s = 0.f;
  for (int c = 0; c < Cin; ++c) s += ir[c] * wr[c];
  out[i] = s;
}

// ---------------- BN over the 16-sample batch axis + lrelu ----------------
__global__ void k_bn_batch(const float* __restrict__ t, const float* __restrict__ g,
                           const float* __restrict__ bta, float* __restrict__ out, int Cout) {
  int o = blockIdx.x * blockDim.x + threadIdx.x;
  if (o >= Cout) return;
  float s1 = 0.f, s2 = 0.f;
  for (int b = 0; b < 16; ++b) { float v = t[b * Cout + o]; s1 += v; s2 += v * v; }
  float mean = s1 * (1.0f / 16.0f);
  float var = s2 * (1.0f / 16.0f) - mean * mean;
  float sc = g[o] * rsqrtf(var + 1e-5f);
  float sh = bta[o] - mean * sc;
  for (int b = 0; b < 16; ++b) out[b * Cout + o] = lrelu(t[b * Cout + o] * sc + sh);
}

// ==================== host side ====================
static void edge_block(hipStream_t stream, const float* xin, float* xout, int O,
                       const float* w, const float* resw,
                       const float* g, const float* bta,
                       const float* para, int pb,
                       float* sqb, int* idxb, float* ybuf, float* rbuf, float* stats) {
  k_sq<<<64, 256, 0, stream>>>(xin, sqb);
  k_knn<<<dim3(64, 16), 128, 0, stream>>>(xin, sqb, idxb);
  (void)hipMemsetAsync(stats, 0, 2 * O * sizeof(float), stream);
  k_edge_gemm<<<10240, 256, 0, stream>>>(xin, idxb, w, para, pb, ybuf, stats, O);
  if (resw)
    k_edge_gemm<<<10240, 256, 0, stream>>>(xin, idxb, resw, para, pb, rbuf, nullptr, O);
  k_bnparams<<<(O + 255) / 256, 256, 0, stream>>>(stats, g, bta, 1.0f / 163840.0f, O);
  int tot = 16 * 1024 * O;
  k_edge_fin<<<(tot + 255) / 256, 256, 0, stream>>>(ybuf, resw ? rbuf : nullptr, stats, xout, O);
}

extern "C" void kernel_launch(void* const* d_in, const int* in_sizes, int n_in,
                              void* d_out, int out_size, void* d_ws, size_t ws_size,
                              hipStream_t stream) {
  (void)in_sizes; (void)n_in; (void)out_size; (void)ws_size;
  const float* disc  = (const float*)d_in[0];
  const float* x     = (const float*)d_in[1];
  /* d_in[2] = label, unused by the reference */
  const float* pos_w = (const float*)d_in[3];
  const float* pos_b = (const float*)d_in[4];
  const float* para  = (const float*)d_in[5];
  const float* w1    = (const float*)d_in[6];
  const float* g1    = (const float*)d_in[7];
  const float* b1    = (const float*)d_in[8];
  const float* res1w = (const float*)d_in[9];
  const float* w2    = (const float*)d_in[10];
  const float* g2    = (const float*)d_in[11];
  const float* b2    = (const float*)d_in[12];
  const float* w3    = (const float*)d_in[13];
  const float* g3    = (const float*)d_in[14];
  const float* b3    = (const float*)d_in[15];
  const float* res3w = (const float*)d_in[16];
  const float* w4    = (const float*)d_in[17];
  const float* g4    = (const float*)d_in[18];
  const float* b4    = (const float*)d_in[19];
  const float* w5    = (const float*)d_in[20];
  const float* g5    = (const float*)d_in[21];
  const float* b5    = (const float*)d_in[22];
  const float* lin1w = (const float*)d_in[23];
  const float* g6    = (const float*)d_in[24];
  const float* b6    = (const float*)d_in[25];
  const float* lin2w = (const float*)d_in[26];
  const float* g7    = (const float*)d_in[27];
  const float* b7    = (const float*)d_in[28];

  char* ws = (char*)d_ws;
  const size_t MB = (size_t)1 << 20;
  float* x0    = (float*)(ws + 0 * MB);    //  8 MB (B,128,N)
  float* x1b   = (float*)(ws + 8 * MB);    //  8 MB
  float* x2b   = (float*)(ws + 16 * MB);   //  8 MB
  float* x3b   = (float*)(ws + 24 * MB);   //  8 MB
  float* x4b   = (float*)(ws + 32 * MB);   // 16 MB (B,256,N)
  float* sqb   = (float*)(ws + 48 * MB);   // 64 KB
  int*   idxb  = (int*)  (ws + 49 * MB);   // 640 KB
  float* ybuf  = (float*)(ws + 50 * MB);   // 160 MB (also reused as y5)
  float* rbuf  = (float*)(ws + 210 * MB);  // 80 MB
  float* stats = (float*)(ws + 290 * MB);  // 8 KB
  float* pbuf  = (float*)(ws + 291 * MB);  // 128 KB
  float* tmp1  = (float*)(ws + 292 * MB);  // 32 KB
  float* h1    = (float*)(ws + 293 * MB);  // 32 KB
  float* tmp2  = (float*)(ws + 294 * MB);  // 16 KB

  k_pos<<<8192, 256, 0, stream>>>(x, disc, pos_w, pos_b, x0);

  edge_block(stream, x0,  x1b, 128, w1, res1w,  g1, b1, para, 0, sqb, idxb, ybuf, rbuf, stats);
  edge_block(stream, x1b, x2b, 128, w2, nullptr, g2, b2, para, 1, sqb, idxb, ybuf, rbuf, stats);
  edge_block(stream, x2b, x3b, 128, w3, res3w,  g3, b3, para, 2, sqb, idxb, ybuf, rbuf, stats);
  edge_block(stream, x3b, x4b, 256, w4, nullptr, g4, b4, para, 3, sqb, idxb, ybuf, rbuf, stats);

  (void)hipMemsetAsync(stats, 0, 2 * 1024 * sizeof(float), stream);
  k_w5_gemm<<<1024, 256, 0, stream>>>(x1b, x2b, x3b, x4b, w5, ybuf, stats);
  k_bnparams<<<4, 256, 0, stream>>>(stats, g5, b5, 1.0f / 16384.0f, 1024);
  k_pool<<<2048, 256, 0, stream>>>(ybuf, stats, pbuf);

  k_lin<<<32, 256, 0, stream>>>(pbuf, lin1w, tmp1, 2048, 512);
  k_bn_batch<<<2, 256, 0, stream>>>(tmp1, g6, b6, h1, 512);
  k_lin<<<16, 256, 0, stream>>>(h1, lin2w, tmp2, 512, 256);
  k_bn_batch<<<1, 256, 0, stream>>>(tmp2, g7, b7, (float*)d_out, 256);
}